// QuantizedEmbeddingBackbone_84636625535420
// MI455X (gfx1250) — compile-verified
//
#include <hip/hip_runtime.h>

typedef __attribute__((ext_vector_type(2))) float v2f;
typedef __attribute__((ext_vector_type(8))) float v8f;

#define BN   16384   // B*N query points
#define NKEY 16384   // K voxel keys
#define DD   32      // embedding dim

// ---------------------------------------------------------------------------
// Kernel 1: pack each key into the WMMA B-operand friendly homogeneous form
//           column_k = (-2kx, -2ky, -2kz, kx^2+ky^2+kz^2)
// so that a single V_WMMA_F32_16X16X4_F32 yields  D[m][n] = -2 p.k + |k|^2
// (== dist - |p|^2, argmin-equivalent to the true squared distance).
// ---------------------------------------------------------------------------
__global__ void pack_keys_kernel(const float* __restrict__ keys,
                                 float* __restrict__ kpack) {
  int i = blockIdx.x * blockDim.x + threadIdx.x;
  if (i >= NKEY) return;
  float kx = keys[3 * i + 0];
  float ky = keys[3 * i + 1];
  float kz = keys[3 * i + 2];
  float4 v = make_float4(-2.0f * kx, -2.0f * ky, -2.0f * kz,
                         kx * kx + ky * ky + kz * kz);
  ((float4*)kpack)[i] = v;
}

// ---------------------------------------------------------------------------
// Kernel 2: pointcloud passthrough (second tuple output of the reference).
// ---------------------------------------------------------------------------
__global__ void copy_pc_kernel(const float* __restrict__ pc,
                               float* __restrict__ out) {
  int i = blockIdx.x * blockDim.x + threadIdx.x;
  if (i < BN * 3) out[i] = pc[i];
}

// ---------------------------------------------------------------------------
// Kernel 3: one wave per 16-point tile. Loop over 1024 key tiles of 16:
//   1 x v_wmma_f32_16x16x4_f32 per tile + 8 running (min,argmin) slots.
// F32 16x16 C/D layout: VGPR r, lanes 0-15 -> (M=r, N=lane); lanes 16-31 ->
// (M=r+8, N=lane-16). A 16x4 f32: lanes 0-15 carry K={0,1}, lanes 16-31 carry
// K={2,3}; B 4x16 mirrors with N across lanes.
// ---------------------------------------------------------------------------
__global__ void __launch_bounds__(256)
nn_argmin_gather_kernel(const float* __restrict__ pc,
                        const float* __restrict__ kpack,
                        const float* __restrict__ emb,
                        float* __restrict__ feats) {
  const int lane = threadIdx.x & 31;
  const int wave = blockIdx.x * (blockDim.x >> 5) + (threadIdx.x >> 5);
  const int tile = wave;            // 16 points per wave
  const int m    = lane & 15;       // row / column within tile for this lane
  const int half = lane >> 4;       // 0 -> K={0,1}, 1 -> K={2,3}

  // --- A fragment: point row (x, y, z, 1) -------------------------------
  const int p = tile * 16 + m;
  const float px = pc[3 * p + 0];
  const float py = pc[3 * p + 1];
  const float pz = pc[3 * p + 2];
  v2f a;
  a.x = half ? pz : px;
  a.y = half ? 1.0f : py;

  const v2f* __restrict__ bp = (const v2f*)kpack;  // 2 x v2f per key

  float best[8];
  int   bidx[8];
#pragma unroll
  for (int r = 0; r < 8; ++r) { best[r] = 3.4e38f; bidx[r] = 0; }

#pragma unroll 4
  for (int kt = 0; kt < NKEY / 16; ++kt) {
    const int j = kt * 16 + m;            // this lane's key column index
    v2f b = bp[2 * j + half];             // global_load_b64, L2-resident
    v8f c = {0.f, 0.f, 0.f, 0.f, 0.f, 0.f, 0.f, 0.f};
    v8f d = __builtin_amdgcn_wmma_f32_16x16x4_f32(
        /*neg_a=*/false, a, /*neg_b=*/false, b,
        /*c_mod=*/(short)0, c, /*reuse_a=*/false, /*reuse_b=*/false);
#pragma unroll
    for (int r = 0; r < 8; ++r) {
      float dv = d[r];
      bool lt  = dv < best[r];
      bidx[r]  = lt ? j  : bidx[r];
      best[r]  = lt ? dv : best[r];
    }
  }

  // --- argmin reduction across the 16 columns (xor within each half) ----
#pragma unroll
  for (int mask = 1; mask <= 8; mask <<= 1) {
#pragma unroll
    for (int r = 0; r < 8; ++r) {
      float ov = __shfl_xor(best[r], mask, 32);
      int   oi = __shfl_xor(bidx[r], mask, 32);
      bool take = (ov < best[r]) || (ov == best[r] && oi < bidx[r]);
      best[r] = take ? ov : best[r];
      bidx[r] = take ? oi : bidx[r];
    }
  }

  // --- embedding gather: row rr's winner lives in lane 0 (rr<8) or lane 16,
  //     slot rr&7; broadcast it and stream emb[idx][0..31] coalesced. -----
#pragma unroll
  for (int rr = 0; rr < 16; ++rr) {
    int idx = __shfl(bidx[rr & 7], (rr < 8) ? 0 : 16, 32);
    feats[(size_t)(tile * 16 + rr) * DD + lane] = emb[(size_t)idx * DD + lane];
  }
}

// ---------------------------------------------------------------------------
extern "C" void kernel_launch(void* const* d_in, const int* in_sizes, int n_in,
                              void* d_out, int out_size, void* d_ws,
                              size_t ws_size, hipStream_t stream) {
  const float* pc   = (const float*)d_in[0];  // (B,N,3)  = 49152 f32
  const float* keys = (const float*)d_in[1];  // (K,3)    = 49152 f32
  const float* emb  = (const float*)d_in[2];  // (K,32)   = 524288 f32
  float* out   = (float*)d_out;               // feats (BN*32) ++ pc (BN*3)
  float* kpack = (float*)d_ws;                // (K,4) packed keys, 256 KB

  // 1) pack keys into homogeneous B-operand form
  pack_keys_kernel<<<NKEY / 256, 256, 0, stream>>>(keys, kpack);

  // 2) pointcloud passthrough into the tail of d_out
  copy_pc_kernel<<<(BN * 3 + 255) / 256, 256, 0, stream>>>(
      pc, out + (size_t)BN * DD);

  // 3) WMMA distance + argmin + gather: 1024 waves -> 128 blocks x 256 thr
  nn_argmin_gather_kernel<<<(BN / 16) * 32 / 256, 256, 0, stream>>>(
      pc, kpack, emb, out);
}